// QuantumRLAgent_19877108645904
// MI455X (gfx1250) — compile-verified
//
#include <hip/hip_runtime.h>
#include <math.h>

#define NQ      18
#define QDIM    (1 << NQ)        // 262144 statevector entries
#define QMASK   (QDIM - 1)       // 0x3FFFF
#define MINJ    (4096 * 18)      // 73728 injected amplitudes
#define BATCH   4096
#define SDIM    256
#define HID     64
#define GEMV_BLOCKS 1024

typedef __attribute__((ext_vector_type(2))) float v2f;
typedef __attribute__((ext_vector_type(8))) float v8f;

// ---------------------------------------------------------------- init
__global__ void k_init(float2* __restrict__ svA) {
  int stride = gridDim.x * blockDim.x;
  for (int n = blockIdx.x * blockDim.x + threadIdx.x; n < QDIM; n += stride) {
    float2 v; v.x = (n == 0) ? 1.f : 0.f; v.y = 0.f;
    svA[n] = v;
  }
}

// ---------------------------------------------------------------- encoder GEMM1 (WMMA f32 16x16x4)
// h = relu(state(4096x256) @ W1(256x64) + b1)   one 16x16 tile per wave
__global__ void __launch_bounds__(256) k_enc1(const float* __restrict__ S,
                                              const float* __restrict__ W1,
                                              const float* __restrict__ b1,
                                              float* __restrict__ H) {
  int lane = threadIdx.x & 31;
  int wave = threadIdx.x >> 5;
  int t  = blockIdx.x * 8 + wave;          // 1024 tiles: 256 M-tiles x 4 N-tiles
  int mt = t >> 2, nt = t & 3;
  int m0 = mt * 16, n0 = nt * 16;
  int mr = lane & 15;                      // A-matrix row (M)
  int kh = (lane >> 4) * 2;                // K sub-offset per ISA layout
  int nc = lane & 15;                      // B/C column (N)
  v8f c = {};
  for (int k0 = 0; k0 < SDIM; k0 += 4) {
    int kb = k0 + kh;
    v2f a, b;
    const float* ap = S + (size_t)(m0 + mr) * SDIM + kb;
    a[0] = ap[0];                          // A[m, kb], A[m, kb+1]
    a[1] = ap[1];
    b[0] = W1[(size_t)kb       * HID + n0 + nc];   // B[kb,   n]
    b[1] = W1[(size_t)(kb + 1) * HID + n0 + nc];   // B[kb+1, n]
    c = __builtin_amdgcn_wmma_f32_16x16x4_f32(false, a, false, b,
                                              (short)0, c, false, false);
  }
  int rbase = (lane >> 4) * 8;             // C/D: v[i] holds rows i (lanes0-15) / i+8 (lanes16-31)
  #pragma unroll
  for (int i = 0; i < 8; ++i) {
    int row = m0 + rbase + i;
    int col = n0 + nc;
    float v = c[i] + b1[col];
    H[(size_t)row * HID + col] = v > 0.f ? v : 0.f;
  }
}

// ---------------------------------------------------------------- encoder GEMM2 + complex injection vector
// enc = h @ W2(64x36) + b2 ; x[row*18+q] = enc[:,q] * exp(i*enc[:,q+18])
__global__ void __launch_bounds__(256) k_enc2(const float* __restrict__ H,
                                              const float* __restrict__ W2,
                                              const float* __restrict__ b2,
                                              float2* __restrict__ X) {
  __shared__ float w[64 * 36];
  for (int i = threadIdx.x; i < 64 * 36; i += blockDim.x) w[i] = W2[i];
  __syncthreads();
  int t = blockIdx.x * blockDim.x + threadIdx.x;   // exactly 73728 threads
  int row = t / 18, q = t - row * 18;
  float aa = b2[q], pp = b2[q + 18];
  const float* hr = H + (size_t)row * HID;
  #pragma unroll 8
  for (int k = 0; k < HID; ++k) {
    float hk = hr[k];
    aa = fmaf(hk, w[k * 36 + q],      aa);
    pp = fmaf(hk, w[k * 36 + q + 18], pp);
  }
  float sp, cp; sincosf(pp, &sp, &cp);
  float2 xv; xv.x = aa * cp; xv.y = aa * sp;
  X[t] = xv;
}

// ---------------------------------------------------------------- fuse Rz*Ry*Rx -> 54 2x2 complex gates
__global__ void k_gates(const float* __restrict__ th, float* __restrict__ G) {
  int t = threadIdx.x;
  if (t >= 54) return;
  float a = th[t * 3 + 0] * 0.5f, b = th[t * 3 + 1] * 0.5f, c = th[t * 3 + 2] * 0.5f;
  float ca = cosf(a), sa = sinf(a);
  float cb = cosf(b), sb = sinf(b);
  float cc = cosf(c), sc = sinf(c);
  // M = Ry * Rx
  float m00r =  cb * ca, m00i =  sb * sa;
  float m01r = -sb * ca, m01i = -cb * sa;
  float m10r =  sb * ca, m10i = -cb * sa;
  float m11r =  cb * ca, m11i = -sb * sa;
  // U = Rz * M : row0 *= e^{-ic/2}, row1 *= e^{+ic/2}
  float* g = G + t * 8;
  g[0] = m00r * cc + m00i * sc;  g[1] = m00i * cc - m00r * sc;
  g[2] = m01r * cc + m01i * sc;  g[3] = m01i * cc - m01r * sc;
  g[4] = m10r * cc - m10i * sc;  g[5] = m10i * cc + m10r * sc;
  g[6] = m11r * cc - m11i * sc;  g[7] = m11i * cc + m11r * sc;
}

// ---------------------------------------------------------------- ||x||^2 (single block, deterministic)
__global__ void k_xnorm(const float2* __restrict__ X, float* __restrict__ out) {
  __shared__ float red[256];
  float s = 0.f;
  for (int i = threadIdx.x; i < MINJ; i += 256) {
    float2 v = X[i];
    s = fmaf(v.x, v.x, fmaf(v.y, v.y, s));
  }
  red[threadIdx.x] = s; __syncthreads();
  for (int off = 128; off > 0; off >>= 1) {
    if (threadIdx.x < off) red[threadIdx.x] += red[threadIdx.x + off];
    __syncthreads();
  }
  if (threadIdx.x == 0) *out = red[0];
}

// ---------------------------------------------------------------- complex 2x2 gate on an LDS pair
__device__ __forceinline__ void apply_pair(float2* ls, int i0, int i1, const float* g) {
  float2 v0 = ls[i0], v1 = ls[i1], r0, r1;
  r0.x = g[0]*v0.x - g[1]*v0.y + g[2]*v1.x - g[3]*v1.y;
  r0.y = g[0]*v0.y + g[1]*v0.x + g[2]*v1.y + g[3]*v1.x;
  r1.x = g[4]*v0.x - g[5]*v0.y + g[6]*v1.x - g[7]*v1.y;
  r1.y = g[4]*v0.y + g[5]*v0.x + g[6]*v1.y + g[7]*v1.x;
  ls[i0] = r0; ls[i1] = r1;
}

// ---------------------------------------------------------------- phase 1: gates on bits 0..11 (qubits 17..6)
// optional gather applies the previous layer's CNOT-ladder permutation: src = n ^ (n<<1)
__global__ void __launch_bounds__(256) k_qlow(const float2* __restrict__ in,
                                              float2* __restrict__ out,
                                              const float* __restrict__ G,
                                              int layer, int perm) {
  __shared__ float2 ls[4096];
  int base = blockIdx.x << 12;             // bits 12..17 fixed per block
  for (int t = 0; t < 16; ++t) {
    int l = threadIdx.x + t * 256;
    int n = base | l;
    int src = perm ? (n ^ ((n << 1) & QMASK)) : n;
    ls[l] = in[src];
  }
  __syncthreads();
  for (int kb = 0; kb < 12; ++kb) {
    const float* g = G + (size_t)(layer * NQ + (NQ - 1 - kb)) * 8;
    float gu[8];
    #pragma unroll
    for (int i = 0; i < 8; ++i) gu[i] = g[i];
    int low = (1 << kb) - 1;
    #pragma unroll
    for (int t = 0; t < 8; ++t) {
      int p  = threadIdx.x + t * 256;       // 2048 pairs
      int i0 = ((p >> kb) << (kb + 1)) | (p & low);
      apply_pair(ls, i0, i0 | (1 << kb), gu);
    }
    __syncthreads();
  }
  for (int t = 0; t < 16; ++t) {
    int l = threadIdx.x + t * 256;
    out[base | l] = ls[l];
  }
}

// ---------------------------------------------------------------- phase 2: gates on bits 12..17 (qubits 5..0)
// block owns bits {0..5} U {12..17}; bits 6..11 = blockIdx. In-place (disjoint sets).
__global__ void __launch_bounds__(256) k_qhigh(float2* __restrict__ buf,
                                               const float* __restrict__ G,
                                               int layer) {
  __shared__ float2 ls[4096];
  int bb = blockIdx.x << 6;
  for (int t = 0; t < 16; ++t) {
    int l = threadIdx.x + t * 256;
    int n = (l & 63) | bb | ((l >> 6) << 12);   // 512B contiguous runs
    ls[l] = buf[n];
  }
  __syncthreads();
  for (int j = 0; j < 6; ++j) {
    int kb = 6 + j;                              // local bit <-> global bit 12+j
    const float* g = G + (size_t)(layer * NQ + (5 - j)) * 8;
    float gu[8];
    #pragma unroll
    for (int i = 0; i < 8; ++i) gu[i] = g[i];
    int low = (1 << kb) - 1;
    #pragma unroll
    for (int t = 0; t < 8; ++t) {
      int p  = threadIdx.x + t * 256;
      int i0 = ((p >> kb) << (kb + 1)) | (p & low);
      apply_pair(ls, i0, i0 | (1 << kb), gu);
    }
    __syncthreads();
  }
  for (int t = 0; t < 16; ++t) {
    int l = threadIdx.x + t * 256;
    int n = (l & 63) | bb | ((l >> 6) << 12);
    buf[n] = ls[l];
  }
}

// ---------------------------------------------------------------- inject normalized x (with final perm gather), |y|^2 partials
__global__ void __launch_bounds__(256) k_inject(const float2* __restrict__ sv,
                                                const float2* __restrict__ X,
                                                const float* __restrict__ xnorm,
                                                float2* __restrict__ Y,
                                                float* __restrict__ part) {
  __shared__ float red[256];
  float xn = *xnorm;
  float rx = xn > 0.f ? rsqrtf(xn) : 0.f;
  float s = 0.f;
  int base = blockIdx.x * 512;
  #pragma unroll
  for (int t = 0; t < 2; ++t) {
    int m   = base + threadIdx.x + t * 256;
    int src = m ^ ((m << 1) & QMASK);        // pending CNOT-ladder permutation
    float2 v = sv[src];
    if (m < MINJ) {
      float2 xv = X[m];
      v.x = fmaf(xv.x, rx, v.x);
      v.y = fmaf(xv.y, rx, v.y);
    }
    Y[m] = v;
    s = fmaf(v.x, v.x, fmaf(v.y, v.y, s));
  }
  red[threadIdx.x] = s; __syncthreads();
  for (int off = 128; off > 0; off >>= 1) {
    if (threadIdx.x < off) red[threadIdx.x] += red[threadIdx.x + off];
    __syncthreads();
  }
  if (threadIdx.x == 0) part[blockIdx.x] = red[0];
}

__global__ void k_ynorm(const float* __restrict__ part, float* __restrict__ out) {
  __shared__ float red[256];
  float s = 0.f;
  for (int i = threadIdx.x; i < 512; i += 256) s += part[i];
  red[threadIdx.x] = s; __syncthreads();
  for (int off = 128; off > 0; off >>= 1) {
    if (threadIdx.x < off) red[threadIdx.x] += red[threadIdx.x + off];
    __syncthreads();
  }
  if (threadIdx.x == 0) *out = red[0];
}

// ---------------------------------------------------------------- bandwidth-dominant GEMVs: cl @ {dec_w1, val_w1}
// blockIdx.y selects matrix. 8 waves/block, each wave owns 64 consecutive rows,
// lane holds 2 columns; rows are 64 contiguous floats -> fully coalesced b64 streams.
__global__ void __launch_bounds__(256) k_gemv(const float2* __restrict__ Y,
                                              const float* __restrict__ Wd,
                                              const float* __restrict__ Wv,
                                              float* __restrict__ part) {
  const float* W = (blockIdx.y == 0) ? Wd : Wv;
  int lane = threadIdx.x & 31, wid = threadIdx.x >> 5;
  const int rowsPerWave = (2 * QDIM) / (GEMV_BLOCKS * 8);   // 64
  int j0 = (blockIdx.x * 8 + wid) * rowsPerWave;            // never crosses the Re/Im boundary
  float a0 = 0.f, a1 = 0.f;
  for (int r = 0; r < rowsPerWave; ++r) {
    int j = j0 + r;
    float clj = (j < QDIM) ? Y[j].x : Y[j - QDIM].y;        // cl = [Re(sv); Im(sv)]
    if (r + 8 < rowsPerWave)
      __builtin_prefetch(W + (size_t)(j + 8) * HID + 2 * lane, 0, 1);
    float2 w = *(const float2*)(W + (size_t)j * HID + 2 * lane);
    a0 = fmaf(clj, w.x, a0);
    a1 = fmaf(clj, w.y, a1);
  }
  __shared__ float red[8][64];
  red[wid][2 * lane]     = a0;
  red[wid][2 * lane + 1] = a1;
  __syncthreads();
  if (threadIdx.x < 64) {
    float s = 0.f;
    #pragma unroll
    for (int w = 0; w < 8; ++w) s += red[w][threadIdx.x];
    part[((size_t)blockIdx.y * GEMV_BLOCKS + blockIdx.x) * 64 + threadIdx.x] = s;
  }
}

__global__ void k_gred(const float* __restrict__ part, float* __restrict__ g) {
  int t = threadIdx.x;                 // 128 = 2 matrices x 64 cols
  int mat = t >> 6, col = t & 63;
  const float* p = part + (size_t)mat * GEMV_BLOCKS * 64 + col;
  float s = 0.f;
  for (int b = 0; b < GEMV_BLOCKS; ++b) s += p[(size_t)b * 64];
  g[t] = s;
}

// ---------------------------------------------------------------- heads: relu(g/||sv|| + b1) @ W2 + b2
__global__ void k_final(const float* __restrict__ g, const float* __restrict__ ynorm,
                        const float* __restrict__ db1, const float* __restrict__ dW2,
                        const float* __restrict__ db2, const float* __restrict__ vb1,
                        const float* __restrict__ vW2, const float* __restrict__ vb2,
                        float* __restrict__ out) {
  __shared__ float hd[64], hv[64];
  float inv = rsqrtf(*ynorm);
  int t = threadIdx.x;
  if (t < 64) {
    float a = g[t]      * inv + db1[t]; hd[t] = a > 0.f ? a : 0.f;
    float b = g[64 + t] * inv + vb1[t]; hv[t] = b > 0.f ? b : 0.f;
  }
  __syncthreads();
  if (t < 18) {
    float s = db2[t];
    for (int k = 0; k < 64; ++k) s = fmaf(hd[k], dW2[k * 18 + t], s);
    out[t] = s;
  } else if (t == 18) {
    float s = vb2[0];
    for (int k = 0; k < 64; ++k) s = fmaf(hv[k], vW2[k], s);
    out[18] = s;
  }
}

// ---------------------------------------------------------------- host launch
extern "C" void kernel_launch(void* const* d_in, const int* in_sizes, int n_in,
                              void* d_out, int out_size, void* d_ws, size_t ws_size,
                              hipStream_t stream) {
  const float* state  = (const float*)d_in[0];
  const float* enc_w1 = (const float*)d_in[1];
  const float* enc_b1 = (const float*)d_in[2];
  const float* enc_w2 = (const float*)d_in[3];
  const float* enc_b2 = (const float*)d_in[4];
  const float* qparam = (const float*)d_in[5];
  const float* dec_w1 = (const float*)d_in[6];
  const float* dec_b1 = (const float*)d_in[7];
  const float* dec_w2 = (const float*)d_in[8];
  const float* dec_b2 = (const float*)d_in[9];
  const float* val_w1 = (const float*)d_in[10];
  const float* val_b1 = (const float*)d_in[11];
  const float* val_w2 = (const float*)d_in[12];
  const float* val_b2 = (const float*)d_in[13];

  float* ws = (float*)d_ws;                       // all offsets in floats, 8B-aligned
  float2* svA  = (float2*)(ws);                   // 2*QDIM
  float2* svB  = (float2*)(ws + 2 * QDIM);        // 2*QDIM
  float2* Y    = (float2*)(ws + 4 * QDIM);        // 2*QDIM
  float*  H    = ws + 6 * QDIM;                   // BATCH*64
  float2* X    = (float2*)(ws + 6 * QDIM + BATCH * HID);  // 2*MINJ
  float*  G    = ws + 6 * QDIM + BATCH * HID + 2 * MINJ;  // 54*8 = 432
  float*  xn   = G + 432;
  float*  yn   = xn + 1;
  float*  partN = yn + 1;                         // 512
  float*  partG = partN + 512;                    // 2*GEMV_BLOCKS*64
  float*  gacc  = partG + 2 * GEMV_BLOCKS * 64;   // 128

  k_init <<<512, 256, 0, stream>>>(svA);
  k_enc1 <<<128, 256, 0, stream>>>(state, enc_w1, enc_b1, H);
  k_enc2 <<<288, 256, 0, stream>>>(H, enc_w2, enc_b2, X);
  k_gates<<<1,   64,  0, stream>>>(qparam, G);
  k_xnorm<<<1,   256, 0, stream>>>(X, xn);

  for (int L = 0; L < 3; ++L) {                   // ping-pong: A->B, B->A, A->B
    const float2* in = (L & 1) ? svB : svA;
    float2*      out = (L & 1) ? svA : svB;
    k_qlow <<<64, 256, 0, stream>>>(in, out, G, L, (L > 0) ? 1 : 0);
    k_qhigh<<<64, 256, 0, stream>>>(out, G, L);
  }

  k_inject<<<512, 256, 0, stream>>>(svB, X, xn, Y, partN);
  k_ynorm <<<1,   256, 0, stream>>>(partN, yn);

  dim3 gg(GEMV_BLOCKS, 2);
  k_gemv <<<gg, 256, 0, stream>>>(Y, dec_w1, val_w1, partG);
  k_gred <<<1, 128, 0, stream>>>(partG, gacc);
  k_final<<<1, 64, 0, stream>>>(gacc, yn, dec_b1, dec_w2, dec_b2,
                                val_b1, val_w2, val_b2, (float*)d_out);
}